// QNN_22823456211604
// MI455X (gfx1250) — compile-verified
//
#include <hip/hip_runtime.h>
#include <hip/hip_bf16.h>
#include <math.h>

// ---------------------------------------------------------------------------
// Problem constants
// ---------------------------------------------------------------------------
#define QDIM 16          // 2^4 basis states
#define QHID 256         // hidden width
#define QB   262144      // batch
#define NB1  128         // blocks for the column-sum reduction
#define U_OFF 2048       // float offset of Ur in ws; Ui at U_OFF+256

typedef float v2f __attribute__((ext_vector_type(2)));
typedef float v4f __attribute__((ext_vector_type(4)));
typedef float v8f __attribute__((ext_vector_type(8)));
typedef unsigned int u32x4 __attribute__((ext_vector_type(4)));
typedef int i32x4 __attribute__((ext_vector_type(4)));
typedef int i32x8 __attribute__((ext_vector_type(8)));

// fp32 WMMA: D(16x16) = A(16x4) * B(4x16) + C   (chain 4x for K=16)
static __device__ __forceinline__ v8f wmma_f32(v2f a, v2f b, v8f c) {
  return __builtin_amdgcn_wmma_f32_16x16x4_f32(false, a, false, b,
                                               (short)0, c, false, false);
}

// intra-wave LDS RAW fence (stores by some lanes -> loads by other lanes)
static __device__ __forceinline__ void lds_fence() {
  asm volatile("s_wait_dscnt 0" ::: "memory");
}
static __device__ __forceinline__ void async_fence() {
  asm volatile("s_wait_asynccnt 0" ::: "memory");
}

// ---------------------------------------------------------------------------
// Kernel 1: per-column sum of x^2 (partial per block) -> ws[NB1*16]
// ---------------------------------------------------------------------------
__global__ void __launch_bounds__(256) k_colsumsq(const float* __restrict__ x,
                                                  float* __restrict__ ws) {
  __shared__ float red[256];
  const int tid = threadIdx.x;
  const int c = tid & 15;        // column 0..15
  const int r = tid >> 4;        // row slot 0..15
  const int rows_per_block = QB / NB1;      // 2048
  const long base = (long)blockIdx.x * rows_per_block;
  float s = 0.f;
  for (int it = r; it < rows_per_block; it += 16) {
    float v = x[(base + it) * QDIM + c];
    s = fmaf(v, v, s);
  }
  red[tid] = s;
  __syncthreads();
  #pragma unroll
  for (int st = 8; st > 0; st >>= 1) {
    if (r < st) red[r * 16 + c] += red[(r + st) * 16 + c];
    __syncthreads();
  }
  if (tid < 16) ws[blockIdx.x * 16 + tid] = red[tid];
}

// ---------------------------------------------------------------------------
// Kernel 2: finalize p_bar, theta; compose full 16x16 complex circuit matrix
// ---------------------------------------------------------------------------
static __device__ void gate_ry(float (*ur)[16], float (*ui)[16], int q, float t) {
  const float c = cosf(0.5f * t), s = sinf(0.5f * t);
  const int p = 1 << (3 - q);
  for (int r0 = 0; r0 < 16; ++r0) {
    if (r0 & p) continue;
    const int r1 = r0 | p;
    for (int cc = 0; cc < 16; ++cc) {
      float a0r = ur[r0][cc], a0i = ui[r0][cc];
      float a1r = ur[r1][cc], a1i = ui[r1][cc];
      ur[r0][cc] = c * a0r - s * a1r;  ui[r0][cc] = c * a0i - s * a1i;
      ur[r1][cc] = s * a0r + c * a1r;  ui[r1][cc] = s * a0i + c * a1i;
    }
  }
}
static __device__ void gate_rz(float (*ur)[16], float (*ui)[16], int q, float t) {
  const float c = cosf(0.5f * t), s = sinf(0.5f * t);
  const int p = 1 << (3 - q);
  for (int r = 0; r < 16; ++r) {
    const float ei = (r & p) ? s : -s;   // e^{-i t/2} / conj
    for (int cc = 0; cc < 16; ++cc) {
      float ar = ur[r][cc], ai = ui[r][cc];
      ur[r][cc] = c * ar - ei * ai;
      ui[r][cc] = c * ai + ei * ar;
    }
  }
}
static __device__ void gate_cnot(float (*ur)[16], float (*ui)[16], int ctrl) {
  const int tgt = (ctrl + 1) & 3;
  const int pc = 1 << (3 - ctrl), pt = 1 << (3 - tgt);
  for (int r = 0; r < 16; ++r) {
    if ((r & pc) && !(r & pt)) {      // transposition r <-> r|pt
      const int r2 = r | pt;
      for (int cc = 0; cc < 16; ++cc) {
        float t0 = ur[r][cc]; ur[r][cc] = ur[r2][cc]; ur[r2][cc] = t0;
        float t1 = ui[r][cc]; ui[r][cc] = ui[r2][cc]; ui[r2][cc] = t1;
      }
    }
  }
}

__global__ void __launch_bounds__(32) k_build_u(const float* __restrict__ partials,
                                                const float* __restrict__ qw,
                                                const float* __restrict__ qgamma,
                                                const float* __restrict__ qbeta,
                                                float* __restrict__ ws) {
  __shared__ float ur[16][16];
  __shared__ float ui[16][16];
  if (threadIdx.x == 0) {
    float pbar[16];
    for (int c = 0; c < 16; ++c) {
      float s = 0.f;
      for (int b = 0; b < NB1; ++b) s += partials[b * 16 + c];
      pbar[c] = s * (1.0f / (float)QB);
    }
    float theta[4];
    for (int i = 0; i < 4; ++i) {
      float m = 0.f;
      for (int k = 0; k < 16; ++k)
        if (((k >> (3 - i)) & 1) == 0) m += pbar[k];     // MASKS @ p_bar
      float v = 1.f - 2.f * m;
      v = fminf(fmaxf(v, -1.f + 1e-6f), 1.f - 1e-6f);
      theta[i] = qgamma[i] * asinf(v) + qbeta[i];
    }
    for (int r = 0; r < 16; ++r)
      for (int c = 0; c < 16; ++c) { ur[r][c] = (r == c) ? 1.f : 0.f; ui[r][c] = 0.f; }
    for (int l = 0; l < 2; ++l) {
      for (int i = 0; i < 4; ++i) {
        gate_ry(ur, ui, i, qw[l * 8 + i * 2 + 0]);
        gate_rz(ur, ui, i, qw[l * 8 + i * 2 + 1]);
      }
      for (int i = 0; i < 4; ++i) gate_cnot(ur, ui, i);
    }
    for (int i = 0; i < 4; ++i) gate_ry(ur, ui, i, theta[i]);
  }
  __syncthreads();
  const float* urf = &ur[0][0];
  const float* uif = &ui[0][0];
  for (int j = 0; j < 16; ++j) {
    int idx = threadIdx.x * 16 + j;
    ws[U_OFF + idx] = (idx < 256) ? urf[idx] : uif[idx - 256];
  }
}

// ---------------------------------------------------------------------------
// Kernel 3: fused  normalize -> |U s|^2 -> relu(P@W1^T+b1) -> @W2^T+b2
// 128 threads = 4 waves; each wave owns a 16-sample tile. 72 WMMAs / wave.
// W1 staged via Tensor Data Mover; x tiles via async global->LDS loads.
// ---------------------------------------------------------------------------
__global__ void __launch_bounds__(128, 2) k_qnn_main(
    const float* __restrict__ x,
    const float* __restrict__ uws,     // ws base (U matrices at U_OFF)
    const float* __restrict__ W1,      // [256,16]
    const float* __restrict__ b1,      // [256]
    const float* __restrict__ W2,      // [2,256]
    const float* __restrict__ b2,      // [2]
    float* __restrict__ out) {         // [B,2]
  __shared__ float sW1[QHID * QDIM];   // 16 KB
  __shared__ float sX[4][QDIM * QDIM];
  __shared__ float sP[4][QDIM * QDIM];
  __shared__ float sNorm[4][QDIM];

  const int tid  = threadIdx.x;
  const int wave = tid >> 5;
  const int lane = tid & 31;
  const int half = lane >> 4;          // 0: lanes 0-15, 1: lanes 16-31
  const int n    = lane & 15;

  // ---- stage W1 with the Tensor Data Mover (one DMA, issued by wave 0) ----
  if (wave == 0) {
    const unsigned long long ga = (unsigned long long)(uintptr_t)W1;
    const unsigned lds = (unsigned)(uintptr_t)&sW1[0];
    u32x4 g0;
    g0[0] = 1u;                                          // count=1 (valid user D#)
    g0[1] = lds;                                         // lds_addr
    g0[2] = (unsigned)(ga & 0xffffffffull);              // global_addr[31:0]
    g0[3] = (unsigned)((ga >> 32) & 0x01ffffffull)       // global_addr[56:32]
          | (2u << 30);                                  // type = 2 ("image")
    i32x8 g1;
    g1[0] = (2 << 16);        // workgroup_mask=0, data_size=2 -> 4 bytes
    g1[1] = (16 << 16);       // tensor_dim0 = 16   (bits 79:48, low half)
    g1[2] = (256 << 16);      // tensor_dim1 = 256  (bits 111:80, low half)
    g1[3] = (16 << 16);       // tile_dim0 = 16     (bits 127:112)
    g1[4] = 256;              // tile_dim1 = 256, tile_dim2 = 0
    g1[5] = 16;               // tensor_dim0_stride = 16 (bits 207:160)
    g1[6] = 0;                // tensor_dim1_stride = 0 (unused for 2D tile)
    g1[7] = 0;
    i32x4 z4 = {0, 0, 0, 0};
#if defined(__clang_major__) && (__clang_major__ >= 23)
    i32x8 z8 = {0, 0, 0, 0, 0, 0, 0, 0};
    __builtin_amdgcn_tensor_load_to_lds(g0, g1, z4, z4, z8, 0);
#else
    __builtin_amdgcn_tensor_load_to_lds(g0, g1, z4, z4, 0);
#endif
    __builtin_amdgcn_s_wait_tensorcnt(0);
  }

  // ---- stage this wave's 16x16 x-tile via async global->LDS b128 ----
  const long tile = (long)blockIdx.x * 4 + wave;
  const long m0 = tile * 16;
  {
    const unsigned long long gbase =
        (unsigned long long)(uintptr_t)(x + m0 * QDIM) + (unsigned)lane * 16u;
    const unsigned ldsa = (unsigned)(uintptr_t)&sX[wave][0] + (unsigned)lane * 16u;
    asm volatile("global_load_async_to_lds_b128 %0, %1, off offset:0"
                 :: "v"(ldsa), "v"(gbase) : "memory");
    asm volatile("global_load_async_to_lds_b128 %0, %1, off offset:512"
                 :: "v"(ldsa), "v"(gbase) : "memory");
  }
  async_fence();

  // ---- per-row inverse L2 norm ----
  if (lane < 16) {
    const v4f* s = (const v4f*)sX[wave];
    v4f a0 = s[lane * 4 + 0], a1 = s[lane * 4 + 1];
    v4f a2 = s[lane * 4 + 2], a3 = s[lane * 4 + 3];
    float ss = a0.x*a0.x + a0.y*a0.y + a0.z*a0.z + a0.w*a0.w
             + a1.x*a1.x + a1.y*a1.y + a1.z*a1.z + a1.w*a1.w
             + a2.x*a2.x + a2.y*a2.y + a2.z*a2.z + a2.w*a2.w
             + a3.x*a3.x + a3.y*a3.y + a3.z*a3.z + a3.w*a3.w;
    sNorm[wave][lane] = rsqrtf(ss);
  }
  __syncthreads();   // W1 (TDM) visible block-wide; norms visible wave-wide

  // ---- A operands: normalized state rows, f32 16x4 layout per K-step ----
  const float scl = sNorm[wave][n];
  v2f a[4];
  #pragma unroll
  for (int t = 0; t < 4; ++t) {
    v2f v = *(const v2f*)&sX[wave][n * 16 + 4 * t + 2 * half];
    a[t].x = v.x * scl;  a[t].y = v.y * scl;
  }
  // ---- B operands: Ur^T / Ui^T, B(k,ncol)=U[ncol][k] (contiguous pairs) ----
  v2f br[4], bi[4];
  #pragma unroll
  for (int t = 0; t < 4; ++t) {
    const int o = n * 16 + 4 * t + 2 * half;
    br[t] = *(const v2f*)&uws[U_OFF + o];
    bi[t] = *(const v2f*)&uws[U_OFF + 256 + o];
  }

  // ---- Yr = S@Ur^T, Yi = S@Ui^T  (8 chained WMMAs) ----
  v8f yr = {0,0,0,0,0,0,0,0}, yi = {0,0,0,0,0,0,0,0};
  #pragma unroll
  for (int t = 0; t < 4; ++t) yr = wmma_f32(a[t], br[t], yr);
  #pragma unroll
  for (int t = 0; t < 4; ++t) yi = wmma_f32(a[t], bi[t], yi);

  // ---- probabilities P = Yr^2 + Yi^2 ----
  v8f p = yr * yr + yi * yi;

  // ---- D-layout -> row-major LDS -> A-layout re-load ----
  #pragma unroll
  for (int v = 0; v < 8; ++v) sP[wave][(v + 8 * half) * 16 + n] = p[v];
  lds_fence();
  v2f pa[4];
  #pragma unroll
  for (int t = 0; t < 4; ++t)
    pa[t] = *(const v2f*)&sP[wave][n * 16 + 4 * t + 2 * half];

  // ---- hidden layer in 16-wide chunks: H = relu(P@W1^T + b1), fold W2 ----
  float ox[8], oy[8];
  #pragma unroll
  for (int v = 0; v < 8; ++v) { ox[v] = 0.f; oy[v] = 0.f; }

  #pragma unroll
  for (int j0 = 0; j0 < QHID; j0 += 16) {
    const int j = j0 + n;
    const float bias = b1[j];
    const float w20 = W2[j];
    const float w21 = W2[QHID + j];
    v8f h = {bias, bias, bias, bias, bias, bias, bias, bias};
    #pragma unroll
    for (int t = 0; t < 4; ++t) {
      v2f bw = *(const v2f*)&sW1[j * 16 + 4 * t + 2 * half];
      h = wmma_f32(pa[t], bw, h);
    }
    #pragma unroll
    for (int v = 0; v < 8; ++v) {
      const float hv = fmaxf(h[v], 0.f);
      ox[v] = fmaf(hv, w20, ox[v]);
      oy[v] = fmaf(hv, w21, oy[v]);
    }
  }

  // ---- reduce partial dot-products across the 16 lanes of each half ----
  #pragma unroll
  for (int m = 1; m < 16; m <<= 1) {
    #pragma unroll
    for (int v = 0; v < 8; ++v) {
      ox[v] += __shfl_xor(ox[v], m, 16);
      oy[v] += __shfl_xor(oy[v], m, 16);
    }
  }

  // ---- epilogue: +b2, write [16,2] tile (lane 0 -> m 0..7, lane 16 -> 8..15)
  if (n == 0) {
    const float c0 = b2[0], c1 = b2[1];
    #pragma unroll
    for (int v = 0; v < 8; ++v) {
      const long mrow = m0 + v + 8 * half;
      v2f o;  o.x = ox[v] + c0;  o.y = oy[v] + c1;
      *(v2f*)&out[mrow * 2] = o;
    }
  }
}

// ---------------------------------------------------------------------------
extern "C" void kernel_launch(void* const* d_in, const int* in_sizes, int n_in,
                              void* d_out, int out_size, void* d_ws, size_t ws_size,
                              hipStream_t stream) {
  (void)in_sizes; (void)n_in; (void)out_size; (void)ws_size;
  const float* x   = (const float*)d_in[0];
  const float* qw  = (const float*)d_in[1];
  const float* qg  = (const float*)d_in[2];
  const float* qb  = (const float*)d_in[3];
  const float* W1  = (const float*)d_in[4];
  const float* b1  = (const float*)d_in[5];
  const float* W2  = (const float*)d_in[6];
  const float* b2  = (const float*)d_in[7];
  float* out = (float*)d_out;
  float* ws  = (float*)d_ws;

  hipLaunchKernelGGL(k_colsumsq, dim3(NB1), dim3(256), 0, stream, x, ws);
  hipLaunchKernelGGL(k_build_u, dim3(1), dim3(32), 0, stream, ws, qw, qg, qb, ws);
  hipLaunchKernelGGL(k_qnn_main, dim3(QB / 64), dim3(128), 0, stream,
                     x, ws, W1, b1, W2, b2, out);
}